// Cluster_Attention_74861279969310
// MI455X (gfx1250) — compile-verified
//
#include <hip/hip_runtime.h>
#include <math.h>

// ---------------------------------------------------------------------------
// Cluster attention: B=16, N=512, C=384, K=64, fp32 end-to-end.
// Stage 1: q = mean(x, axis=1); qn = l2norm(q)
// Stage 2: attn = softmax(qn @ l2norm_cols(centers))
// Stage 3: tm[b] = sum_k attn[b,k] * tran_ms[k]     (memory-bound stream)
// Stage 4: y[b]  = tm[b] @ x[b]                     (WMMA f32, async LDS pipe)
// Stage 5: out   = y @ proj_w^T + proj_b            (WMMA f32, async LDS pipe)
// ---------------------------------------------------------------------------

typedef float v2f __attribute__((ext_vector_type(2)));
typedef float v8f __attribute__((ext_vector_type(8)));

#define Bdim 16
#define Ndim 512
#define Cdim 384
#define Kdim 64
#define NN   (Ndim * Ndim)   // 262144

__device__ __forceinline__ v8f wmma_f32(v2f a, v2f b, v8f c) {
  // 8 args: (neg_a, A, neg_b, B, c_mod, C, reuse_a, reuse_b)
  return __builtin_amdgcn_wmma_f32_16x16x4_f32(false, a, false, b,
                                               (short)0, c, false, false);
}

// --- CDNA5 async global->LDS copies (ASYNCcnt path), GV addressing mode ----
// Generic LDS pointer low 32 bits == wave-relative LDS byte offset (ISA §10.2).
__device__ __forceinline__ uint32_t lds_addr(const void* p) {
  return (uint32_t)(uintptr_t)p;
}
__device__ __forceinline__ void async_load_b32(uint32_t lds, const void* g) {
  asm volatile("global_load_async_to_lds_b32 %0, %1, off"
               :: "v"(lds), "v"(g) : "memory");
}
__device__ __forceinline__ void async_load_b64(uint32_t lds, const void* g) {
  asm volatile("global_load_async_to_lds_b64 %0, %1, off"
               :: "v"(lds), "v"(g) : "memory");
}
#define WAIT_ASYNCCNT(n) asm volatile("s_wait_asynccnt %0" :: "n"(n) : "memory")

// ---------------------------------------------------------------------------
// Kernel 1: per-batch mean over N, then L2-normalize over C.
// grid = B, block = C (384 threads = 12 waves)
// ---------------------------------------------------------------------------
__global__ __launch_bounds__(Cdim)
void k_meanq(const float* __restrict__ x, float* __restrict__ qn) {
  const int b = blockIdx.x;
  const int c = threadIdx.x;
  const float* xb = x + (size_t)b * Ndim * Cdim;

  float s = 0.f;
  for (int n = 0; n < Ndim; ++n) s += xb[(size_t)n * Cdim + c];
  const float q = s * (1.0f / (float)Ndim);

  __shared__ float red[Cdim];
  red[c] = q * q;
  __syncthreads();
  if (c < 128) red[c] += red[c + 128] + red[c + 256];
  __syncthreads();
  for (int off = 64; off > 0; off >>= 1) {
    if (c < off) red[c] += red[c + off];
    __syncthreads();
  }
  const float inv = 1.0f / fmaxf(sqrtf(red[0]), 1e-12f);
  qn[b * Cdim + c] = q * inv;
}

// ---------------------------------------------------------------------------
// Kernel 2: routing weights attn = softmax(qn @ l2norm_cols(centers)).
// grid = B, block = K (64 threads = 2 waves)
// ---------------------------------------------------------------------------
__global__ __launch_bounds__(Kdim)
void k_attn(const float* __restrict__ qn, const float* __restrict__ centers,
            float* __restrict__ attn) {
  const int b = blockIdx.x;
  const int k = threadIdx.x;

  float cn2 = 0.f, dot = 0.f;
  const float* qb = qn + b * Cdim;
  for (int c = 0; c < Cdim; ++c) {
    const float v = centers[c * Kdim + k];   // coalesced over k
    cn2 += v * v;
    dot += qb[c] * v;
  }
  const float logit = dot / fmaxf(sqrtf(cn2), 1e-12f);

  __shared__ float sh[Kdim];
  sh[k] = logit;
  __syncthreads();
  for (int off = 32; off > 0; off >>= 1) {
    if (k < off) sh[k] = fmaxf(sh[k], sh[k + off]);
    __syncthreads();
  }
  const float m = sh[0];
  __syncthreads();
  const float e = expf(logit - m);
  sh[k] = e;
  __syncthreads();
  for (int off = 32; off > 0; off >>= 1) {
    if (k < off) sh[k] += sh[k + off];
    __syncthreads();
  }
  attn[b * Kdim + k] = e / sh[0];
}

// ---------------------------------------------------------------------------
// Kernel 3: tm[b][j] = sum_k attn[b][k] * tran_ms[k][j].
// Memory-bound stream: tran_ms read exactly once (67 MB), 16 FMAs per load.
// grid = NN/256, block = 256
// ---------------------------------------------------------------------------
__global__ __launch_bounds__(256)
void k_tm(const float* __restrict__ attn, const float* __restrict__ tran,
          float* __restrict__ tm) {
  __shared__ float attn_s[Bdim * Kdim];
  for (int i = threadIdx.x; i < Bdim * Kdim; i += 256) attn_s[i] = attn[i];
  __syncthreads();

  const size_t j = (size_t)blockIdx.x * 256 + threadIdx.x;
  float acc[Bdim];
#pragma unroll
  for (int b = 0; b < Bdim; ++b) acc[b] = 0.f;

  for (int k = 0; k < Kdim; ++k) {
    const float t = tran[(size_t)k * NN + j];
#pragma unroll
    for (int b = 0; b < Bdim; ++b) acc[b] = fmaf(attn_s[b * Kdim + k], t, acc[b]);
  }
#pragma unroll
  for (int b = 0; b < Bdim; ++b) tm[(size_t)b * NN + j] = acc[b];
}

// ---------------------------------------------------------------------------
// Kernel 4: y[b] = tm[b] (512x512) @ x[b] (512x384), fp32 WMMA 16x16x4.
// 128 threads (4 waves), block tile 64x64, K-step 32.
// Double-buffered LDS fed by async global->LDS copies; the x-tile transpose
// (BsT[n][k] = x[k][n]) is done by per-lane async B32 scatter.
// grid = (Cdim/64, Ndim/64, B)
// ---------------------------------------------------------------------------
#define LDK 34  // 32 + 2 pad: 8B-aligned float2 rows, conflict-free stride

__global__ __launch_bounds__(128)
void k_bmm(const float* __restrict__ tm, const float* __restrict__ x,
           float* __restrict__ y) {
  const int b    = blockIdx.z;
  const int row0 = blockIdx.y * 64;
  const int col0 = blockIdx.x * 64;
  const int tid  = threadIdx.x;
  const int lane = tid & 31;
  const int wave = tid >> 5;
  const int wRow = (wave >> 1) * 32;
  const int wCol = (wave & 1) * 32;
  const int lm   = lane & 15;
  const int hi   = lane >> 4;

  __shared__ __align__(16) float As[2][64 * LDK];   // A tile, [m][k]
  __shared__ __align__(16) float BsT[2][64 * LDK];  // B tile, transposed [n][k]

  const float* A  = tm + (size_t)b * NN;
  const float* Bx = x  + (size_t)b * Ndim * Cdim;

  v8f acc[2][2] = {};

  // 8 x b64 (A rows) + 16 x b32 (B transpose scatter) = 24 async ops per lane
  auto issue = [&](int buf, int k0) {
#pragma unroll
    for (int j = 0; j < 8; ++j) {
      const int f = tid + j * 128;          // float2 index within 64x32 tile
      const int r = f >> 4, c = (f & 15) * 2;
      async_load_b64(lds_addr(&As[buf][r * LDK + c]),
                     &A[(size_t)(row0 + r) * Ndim + k0 + c]);
    }
#pragma unroll
    for (int j = 0; j < 16; ++j) {
      const int idx = tid + j * 128;        // element index within 32x64 tile
      const int kk = idx >> 6, n = idx & 63;
      async_load_b32(lds_addr(&BsT[buf][n * LDK + kk]),
                     &Bx[(size_t)(k0 + kk) * Cdim + col0 + n]);
    }
  };

  issue(0, 0);
  const int NSTEPS = Ndim / 32;
  for (int ks = 0; ks < NSTEPS; ++ks) {
    const int cur = ks & 1;
    if (ks + 1 < NSTEPS) {
      issue(cur ^ 1, (ks + 1) * 32);
      WAIT_ASYNCCNT(24);                    // in-order: current tile complete
    } else {
      WAIT_ASYNCCNT(0);
    }
    __syncthreads();

#pragma unroll
    for (int kk = 0; kk < 32; kk += 4) {
      const int ak = kk + 2 * hi;  // VGPR0:K+0/K+2, VGPR1:K+1/K+3 per half-wave
#pragma unroll
      for (int tr = 0; tr < 2; ++tr) {
        const v2f a = *(const v2f*)&As[cur][(wRow + tr * 16 + lm) * LDK + ak];
#pragma unroll
        for (int tc = 0; tc < 2; ++tc) {
          const v2f bb = *(const v2f*)&BsT[cur][(wCol + tc * 16 + lm) * LDK + ak];
          acc[tr][tc] = wmma_f32(a, bb, acc[tr][tc]);
        }
      }
    }
    __syncthreads();   // protects buffer reuse (WAR) two steps later
  }

  float* Y = y + (size_t)b * Ndim * Cdim;
#pragma unroll
  for (int tr = 0; tr < 2; ++tr)
#pragma unroll
    for (int tc = 0; tc < 2; ++tc)
#pragma unroll
      for (int v = 0; v < 8; ++v) {
        const int row = row0 + wRow + tr * 16 + v + 8 * hi;  // C/D layout
        const int col = col0 + wCol + tc * 16 + lm;
        Y[(size_t)row * Cdim + col] = acc[tr][tc][v];
      }
}

// ---------------------------------------------------------------------------
// Kernel 5: out = y (8192x384) @ proj_w^T (384x384) + proj_b.
// B^T[n][k] = proj_w[n][k] is proj_w's native row-major layout -> contiguous
// async b64 copies for both tiles.
// grid = (Cdim/64, 8192/64)
// ---------------------------------------------------------------------------
__global__ __launch_bounds__(128)
void k_linear(const float* __restrict__ y, const float* __restrict__ pw,
              const float* __restrict__ pb, float* __restrict__ out) {
  const int row0 = blockIdx.y * 64;
  const int col0 = blockIdx.x * 64;
  const int tid  = threadIdx.x;
  const int lane = tid & 31;
  const int wave = tid >> 5;
  const int wRow = (wave >> 1) * 32;
  const int wCol = (wave & 1) * 32;
  const int lm   = lane & 15;
  const int hi   = lane >> 4;

  __shared__ __align__(16) float As[2][64 * LDK];
  __shared__ __align__(16) float BsT[2][64 * LDK];

  v8f acc[2][2] = {};

  // 8 x b64 (A) + 8 x b64 (proj_w rows) = 16 async ops per lane
  auto issue = [&](int buf, int k0) {
#pragma unroll
    for (int j = 0; j < 8; ++j) {
      const int f = tid + j * 128;
      const int r = f >> 4, c = (f & 15) * 2;
      async_load_b64(lds_addr(&As[buf][r * LDK + c]),
                     &y[(size_t)(row0 + r) * Cdim + k0 + c]);
    }
#pragma unroll
    for (int j = 0; j < 8; ++j) {
      const int f = tid + j * 128;
      const int n = f >> 4, kk = (f & 15) * 2;
      async_load_b64(lds_addr(&BsT[buf][n * LDK + kk]),
                     &pw[(size_t)(col0 + n) * Cdim + k0 + kk]);
    }
  };

  issue(0, 0);
  const int NSTEPS = Cdim / 32;
  for (int ks = 0; ks < NSTEPS; ++ks) {
    const int cur = ks & 1;
    if (ks + 1 < NSTEPS) {
      issue(cur ^ 1, (ks + 1) * 32);
      WAIT_ASYNCCNT(16);
    } else {
      WAIT_ASYNCCNT(0);
    }
    __syncthreads();

#pragma unroll
    for (int kk = 0; kk < 32; kk += 4) {
      const int ak = kk + 2 * hi;
#pragma unroll
      for (int tr = 0; tr < 2; ++tr) {
        const v2f a = *(const v2f*)&As[cur][(wRow + tr * 16 + lm) * LDK + ak];
#pragma unroll
        for (int tc = 0; tc < 2; ++tc) {
          const v2f bb = *(const v2f*)&BsT[cur][(wCol + tc * 16 + lm) * LDK + ak];
          acc[tr][tc] = wmma_f32(a, bb, acc[tr][tc]);
        }
      }
    }
    __syncthreads();
  }

#pragma unroll
  for (int tr = 0; tr < 2; ++tr)
#pragma unroll
    for (int tc = 0; tc < 2; ++tc) {
      const int col = col0 + wCol + tc * 16 + lm;
      const float bias = pb[col];
#pragma unroll
      for (int v = 0; v < 8; ++v) {
        const int row = row0 + wRow + tr * 16 + v + 8 * hi;
        out[(size_t)row * Cdim + col] = acc[tr][tc][v] + bias;
      }
    }
}

// ---------------------------------------------------------------------------
extern "C" void kernel_launch(void* const* d_in, const int* in_sizes, int n_in,
                              void* d_out, int out_size, void* d_ws, size_t ws_size,
                              hipStream_t stream) {
  const float* x       = (const float*)d_in[0];  // [16,512,384]
  const float* centers = (const float*)d_in[1];  // [384,64]
  const float* tran    = (const float*)d_in[2];  // [64,512,512]
  const float* pw      = (const float*)d_in[3];  // [384,384]
  const float* pb      = (const float*)d_in[4];  // [384]
  float* out = (float*)d_out;                    // [16,512,384]

  float* ws   = (float*)d_ws;
  float* qn   = ws;                              // 16*384      = 6144
  float* attn = ws + 6144;                       // 16*64       = 1024
  float* tm   = ws + 8192;                       // 16*512*512  = 4194304
  float* yv   = ws + 8192 + (size_t)Bdim * NN;   // 16*512*384  = 3145728

  k_meanq<<<Bdim, Cdim, 0, stream>>>(x, qn);
  k_attn<<<Bdim, Kdim, 0, stream>>>(qn, centers, attn);
  k_tm<<<NN / 256, 256, 0, stream>>>(attn, tran, tm);

  dim3 gBmm(Cdim / 64, Ndim / 64, Bdim);         // (6, 8, 16)
  k_bmm<<<gBmm, 128, 0, stream>>>(tm, x, yv);

  dim3 gLin(Cdim / 64, (Bdim * Ndim) / 64);      // (6, 128)
  k_linear<<<gLin, 128, 0, stream>>>(yv, pw, pb, out);
}